// GATLayer_47888885350733
// MI455X (gfx1250) — compile-verified
//
#include <hip/hip_runtime.h>

// ---------------------------------------------------------------------------
// GAT layer for MI455X (gfx1250, wave32).
//   B=8, M=1024, F=256, H=8, E=32
// ws layout (floats):
//   [0]                 Wh   : B*H*M*E   = 2,097,152 floats (8 MB)
//   [WH_N]              s1   : B*H*M     = 65,536
//   [WH_N +   S_N]      s2   : 65,536
//   [WH_N + 2*S_N]      rmax : 65,536
//   [WH_N + 3*S_N]      rsum : 65,536
// total ~9 MB of workspace.
//
// Each wave owns TWO 16-row M-tiles so every B fragment (weight / Wh) feeds
// two v_wmma_f32_16x16x4_f32 A-tiles: halves B-load traffic per WMMA on the
// dominant 4.3-GFLOP attention GEMM.
// ---------------------------------------------------------------------------

typedef __attribute__((ext_vector_type(2))) float v2f;
typedef __attribute__((ext_vector_type(8))) float v8f;

namespace gat {
constexpr int Bc = 8, Mc = 1024, Fc = 256, Hc = 8, Ec = 32;
constexpr size_t WH_N = (size_t)Bc * Hc * Mc * Ec;   // Wh elements
constexpr size_t S_N  = (size_t)Bc * Hc * Mc;        // per-row stats
constexpr float LALPHA  = 0.2f;
constexpr float NEG_INF = -9.0e15f;
}

// ---------------------------------------------------------------------------
// Kernel 1: Wh[b,h,m,e] = sum_f x[b,m,f] * W[h,f,e]   (fp32 WMMA 16x16x4)
// plus s1[b,h,m] = Wh . a1[h],  s2[b,h,m] = Wh . a2[h]
// One wave per 32-row slab (two 16-row A tiles sharing B fragments).
// A-frag layout (16x4 f32): lane L holds M=L%16, K = 2*(L/16)+{0,1} in v0,v1.
// B-frag layout (4x16 f32): lane L holds N=L%16, K = 2*(L/16)+{0,1} in v0,v1.
// C/D layout (16x16 f32): lane L holds N=L%16, M = r + 8*(L/16) in VGPR r.
// ---------------------------------------------------------------------------
__global__ __launch_bounds__(32)
void gat_wh_kernel(const float* __restrict__ x,
                   const float* __restrict__ w,
                   const float* __restrict__ att,
                   float* __restrict__ wh,
                   float* __restrict__ s1,
                   float* __restrict__ s2) {
    using namespace gat;
    const int lane = threadIdx.x;           // 0..31
    const int bid  = blockIdx.x;            // b*H*32 + h*32 + mb
    const int mb   = bid & 31;              // 32-row slab index
    const int h    = (bid >> 5) & 7;
    const int b    = bid >> 8;

    const int col  = lane & 15;             // N (or M for A-frag rows)
    const int half = lane >> 4;             // 0/1
    const int k0   = half * 2;              // K pair base inside a K=4 step

    const int m0 = mb * 32;
    const float* xrow0 = x + ((size_t)b * Mc + (m0 + col)) * Fc;       // tile 0
    const float* xrow1 = xrow0 + (size_t)16 * Fc;                       // tile 1
    const float* wbase = w + (size_t)h * Fc * Ec;

    v8f acc[2][2] = {};   // [m-tile][e-tile]
    for (int kk = 0; kk < Fc; kk += 4) {
        v2f af0 = *(const v2f*)(xrow0 + kk + k0);
        v2f af1 = *(const v2f*)(xrow1 + kk + k0);
        const float* wp = wbase + (size_t)(kk + k0) * Ec;
        v2f bf0 = { wp[col],      wp[Ec + col]      };   // e tile 0
        v2f bf1 = { wp[16 + col], wp[Ec + 16 + col] };   // e tile 1
        acc[0][0] = __builtin_amdgcn_wmma_f32_16x16x4_f32(false, af0, false, bf0,
                                                          (short)0, acc[0][0], false, false);
        acc[0][1] = __builtin_amdgcn_wmma_f32_16x16x4_f32(false, af0, false, bf1,
                                                          (short)0, acc[0][1], false, false);
        acc[1][0] = __builtin_amdgcn_wmma_f32_16x16x4_f32(false, af1, false, bf0,
                                                          (short)0, acc[1][0], false, false);
        acc[1][1] = __builtin_amdgcn_wmma_f32_16x16x4_f32(false, af1, false, bf1,
                                                          (short)0, acc[1][1], false, false);
    }

    // att: (H, 2E, 1) flat; a1 = att[h, :E], a2 = att[h, E:]
    const float a1_0 = att[h * 2 * Ec + col];
    const float a1_1 = att[h * 2 * Ec + 16 + col];
    const float a2_0 = att[h * 2 * Ec + Ec + col];
    const float a2_1 = att[h * 2 * Ec + Ec + 16 + col];

    const int bh = b * Hc + h;
    float* whbase = wh + ((size_t)bh * Mc + m0) * Ec;

    #pragma unroll
    for (int t = 0; t < 2; ++t) {
        #pragma unroll
        for (int r = 0; r < 8; ++r) {
            const int m_l = t * 16 + r + 8 * half;   // row within the 32-row slab
            const float v0 = acc[t][0][r];
            const float v1 = acc[t][1][r];
            whbase[(size_t)m_l * Ec + col]      = v0;
            whbase[(size_t)m_l * Ec + 16 + col] = v1;

            // Row dot-products with a1/a2: reduce across the 16 lanes of this row.
            float p1 = v0 * a1_0 + v1 * a1_1;
            float p2 = v0 * a2_0 + v1 * a2_1;
            #pragma unroll
            for (int d = 1; d < 16; d <<= 1) {
                p1 += __shfl_xor(p1, d, 32);
                p2 += __shfl_xor(p2, d, 32);
            }
            if (col == 0) {
                s1[(size_t)bh * Mc + m0 + m_l] = p1;
                s2[(size_t)bh * Mc + m0 + m_l] = p2;
            }
        }
    }
}

// ---------------------------------------------------------------------------
// Kernel 2: per-row softmax stats over the masked leaky scores.
//   e[m,n] = leaky(s1[m]+s2[n]); masked to NEG_INF where adj==0.
//   rmax = max_n e ; rsum = sum_n exp(e - rmax)   (finite-mask semantics,
//   matches reference exactly incl. all-masked -> uniform rows).
// One wave per (b,h,m) row; lanes stride over n (coalesced 128B).
// ---------------------------------------------------------------------------
__global__ __launch_bounds__(256)
void gat_stats_kernel(const int* __restrict__ adj,
                      const float* __restrict__ s1,
                      const float* __restrict__ s2,
                      float* __restrict__ rmax,
                      float* __restrict__ rsum) {
    using namespace gat;
    const int lane = threadIdx.x & 31;
    const int wid  = blockIdx.x * 8 + (threadIdx.x >> 5);   // 0 .. B*H*M-1
    const int m    = wid & (Mc - 1);
    const int bh   = wid >> 10;
    const int b    = bh >> 3;

    const float s1v = s1[(size_t)bh * Mc + m];
    const int*   arow  = adj + ((size_t)b * Mc + m) * Mc;
    const float* s2row = s2 + (size_t)bh * Mc;

    float mx = NEG_INF;
    for (int n = lane; n < Mc; n += 32) {
        float sc = s1v + s2row[n];
        sc = sc > 0.0f ? sc : LALPHA * sc;
        sc = arow[n] > 0 ? sc : NEG_INF;
        mx = fmaxf(mx, sc);
    }
    #pragma unroll
    for (int d = 1; d < 32; d <<= 1) mx = fmaxf(mx, __shfl_xor(mx, d, 32));

    float sum = 0.0f;
    for (int n = lane; n < Mc; n += 32) {
        float sc = s1v + s2row[n];
        sc = sc > 0.0f ? sc : LALPHA * sc;
        sc = arow[n] > 0 ? sc : NEG_INF;
        sum += __expf(sc - mx);
    }
    #pragma unroll
    for (int d = 1; d < 32; d <<= 1) sum += __shfl_xor(sum, d, 32);

    if (lane == 0) {
        rmax[wid] = mx;
        rsum[wid] = sum;
    }
}

// ---------------------------------------------------------------------------
// Kernel 3: fused attention GEMM + ELU + transposed store.
//   out[b,m,h*E+e] = elu( sum_n P[m,n] * Wh[b,h,n,e] ),
//   P = exp(masked_leaky(s1+s2) - rmax) / rsum    (never materialized).
// One wave per 32-row slab (two 16-row P tiles share every Wh B fragment);
// P tiles bounced through LDS to re-layout into WMMA A fragments;
// 16x v_wmma_f32_16x16x4_f32 per 16-column step.
// ---------------------------------------------------------------------------
__global__ __launch_bounds__(32)
void gat_attn_kernel(const int* __restrict__ adj,
                     const float* __restrict__ wh,
                     const float* __restrict__ s1,
                     const float* __restrict__ s2,
                     const float* __restrict__ rmax,
                     const float* __restrict__ rsum,
                     float* __restrict__ out) {
    using namespace gat;
    __shared__ float pblk[2][16 * 16];

    const int lane = threadIdx.x;
    const int bid  = blockIdx.x;
    const int mb   = bid & 31;
    const int h    = (bid >> 5) & 7;
    const int b    = bid >> 8;
    const int bh   = b * Hc + h;

    const int col  = lane & 15;
    const int half = lane >> 4;
    const int k0   = half * 2;
    const int m0   = mb * 32;

    // Per-lane stats for the 16 rows this lane owns (8 per M-tile).
    float s1v[16], rmx[16], rin[16];
    #pragma unroll
    for (int t = 0; t < 2; ++t) {
        #pragma unroll
        for (int r = 0; r < 8; ++r) {
            const int m = m0 + t * 16 + r + 8 * half;
            s1v[t * 8 + r] = s1[(size_t)bh * Mc + m];
            rmx[t * 8 + r] = rmax[(size_t)bh * Mc + m];
            rin[t * 8 + r] = 1.0f / rsum[(size_t)bh * Mc + m];
        }
    }

    const float* whb  = wh + (size_t)bh * Mc * Ec;
    const int*   adjb = adj + (size_t)b * Mc * Mc;
    const float* s2b  = s2 + (size_t)bh * Mc;

    v8f acc[2][2] = {};   // [m-tile][e-tile]
    for (int nb = 0; nb < Mc / 16; ++nb) {
        const int n0 = nb * 16;
        const float s2v = s2b[n0 + col];

        if (nb + 1 < Mc / 16) {   // prefetch next adj tiles (global_prefetch_b8)
            __builtin_prefetch(&adjb[(size_t)(m0 + 8 * half) * Mc + n0 + 16 + col], 0, 0);
            __builtin_prefetch(&adjb[(size_t)(m0 + 16 + 8 * half) * Mc + n0 + 16 + col], 0, 0);
        }

        // Compute both P tiles in C/D layout, stage to LDS row-major [m][n].
        #pragma unroll
        for (int t = 0; t < 2; ++t) {
            #pragma unroll
            for (int r = 0; r < 8; ++r) {
                const int mL = r + 8 * half;
                float sc = s1v[t * 8 + r] + s2v;
                sc = sc > 0.0f ? sc : LALPHA * sc;
                const int a = adjb[(size_t)(m0 + t * 16 + mL) * Mc + n0 + col];
                sc = a > 0 ? sc : NEG_INF;
                pblk[t][mL * 16 + col] = __expf(sc - rmx[t * 8 + r]) * rin[t * 8 + r];
            }
        }
        __syncthreads();   // single wave: lowers to waitcnt (+ S_NOP barrier)

        // 4 K-steps of 4: A = P tiles (from LDS), B = Wh rows n0..n0+15 (shared).
        #pragma unroll
        for (int ks = 0; ks < 4; ++ks) {
            v2f af0 = *(const v2f*)&pblk[0][col * 16 + ks * 4 + k0];
            v2f af1 = *(const v2f*)&pblk[1][col * 16 + ks * 4 + k0];
            const float* wp = whb + (size_t)(n0 + ks * 4 + k0) * Ec;
            v2f bf0 = { wp[col],      wp[Ec + col]      };
            v2f bf1 = { wp[16 + col], wp[Ec + 16 + col] };
            acc[0][0] = __builtin_amdgcn_wmma_f32_16x16x4_f32(false, af0, false, bf0,
                                                              (short)0, acc[0][0], false, false);
            acc[0][1] = __builtin_amdgcn_wmma_f32_16x16x4_f32(false, af0, false, bf1,
                                                              (short)0, acc[0][1], false, false);
            acc[1][0] = __builtin_amdgcn_wmma_f32_16x16x4_f32(false, af1, false, bf0,
                                                              (short)0, acc[1][0], false, false);
            acc[1][1] = __builtin_amdgcn_wmma_f32_16x16x4_f32(false, af1, false, bf1,
                                                              (short)0, acc[1][1], false, false);
        }
        __syncthreads();
    }

    // ELU (alpha=1) + transposed store: out[b, m, h*E + e]
    float* ob = out + (size_t)b * Mc * (Hc * Ec) + (size_t)h * Ec;
    #pragma unroll
    for (int t = 0; t < 2; ++t) {
        #pragma unroll
        for (int r = 0; r < 8; ++r) {
            const int m = m0 + t * 16 + r + 8 * half;
            float v0 = acc[t][0][r];
            float v1 = acc[t][1][r];
            v0 = v0 > 0.0f ? v0 : (__expf(v0) - 1.0f);
            v1 = v1 > 0.0f ? v1 : (__expf(v1) - 1.0f);
            ob[(size_t)m * (Hc * Ec) + col]      = v0;
            ob[(size_t)m * (Hc * Ec) + 16 + col] = v1;
        }
    }
}

// ---------------------------------------------------------------------------
extern "C" void kernel_launch(void* const* d_in, const int* in_sizes, int n_in,
                              void* d_out, int out_size, void* d_ws, size_t ws_size,
                              hipStream_t stream) {
    using namespace gat;
    const float* x   = (const float*)d_in[0];   // (B, M, F) f32
    const int*   adj = (const int*)  d_in[1];   // (B, M, M) i32
    const float* w   = (const float*)d_in[2];   // (H, F, E) f32
    const float* att = (const float*)d_in[3];   // (H, 2E, 1) f32
    float* out = (float*)d_out;                 // (B, M, H*E) f32

    float* ws   = (float*)d_ws;                 // needs ~9 MB
    float* wh   = ws;
    float* s1   = ws + WH_N;
    float* s2   = ws + WH_N + S_N;
    float* rmx  = ws + WH_N + 2 * S_N;
    float* rsm  = ws + WH_N + 3 * S_N;

    const int slabs = Bc * Hc * (Mc / 32);      // 2048 single-wave blocks

    gat_wh_kernel<<<slabs, 32, 0, stream>>>(x, w, att, wh, s1, s2);
    gat_stats_kernel<<<(int)(S_N / 8), 256, 0, stream>>>(adj, s1, s2, rmx, rsm);
    gat_attn_kernel<<<slabs, 32, 0, stream>>>(adj, wh, s1, s2, rmx, rsm, out);
}